// FusionMethodC_46703474376900
// MI455X (gfx1250) — compile-verified
//
#include <hip/hip_runtime.h>
#include <cstdint>

typedef float v2f __attribute__((ext_vector_type(2)));
typedef float v8f __attribute__((ext_vector_type(8)));

#define N_NODES 50000
#define N_EDGES 1600000
#define D_INF   128
#define D_HID   64
#define BN_EPS  1e-5f
#define N_TILES (N_NODES / 16)   // 3125, exact

// ---------------------------------------------------------------------------
// init: deg = 1.0 (self-loop weight), zero BN stats (2 layers x {sum,sumsq})
// ---------------------------------------------------------------------------
__global__ __launch_bounds__(256) void k_init(float* deg, float* stats) {
    int i = blockIdx.x * blockDim.x + threadIdx.x;
    if (i < N_NODES) deg[i] = 1.0f;
    if (i < 4 * D_HID) stats[i] = 0.0f;
}

// ---------------------------------------------------------------------------
// w = sigmoid(alpha)*wsc + (1-sigmoid(alpha))*wfc ; deg[dst] += w   (fp atomics)
// ---------------------------------------------------------------------------
__global__ __launch_bounds__(256) void k_mix_deg(const float* __restrict__ wsc,
                                                 const float* __restrict__ wfc,
                                                 const float* __restrict__ alpha_p,
                                                 const long long* __restrict__ dst,
                                                 float* __restrict__ w,
                                                 float* __restrict__ deg) {
    int e = blockIdx.x * blockDim.x + threadIdx.x;
    if (e >= N_EDGES) return;
    float a  = 1.0f / (1.0f + __expf(-alpha_p[0]));
    float we = a * wsc[e] + (1.0f - a) * wfc[e];
    w[e] = we;
    unsafeAtomicAdd(&deg[(int)dst[e]], we);
}

__global__ __launch_bounds__(256) void k_dinv(const float* __restrict__ deg,
                                              float* __restrict__ dinv) {
    int i = blockIdx.x * blockDim.x + threadIdx.x;
    if (i >= N_NODES) return;
    float d = deg[i];
    dinv[i] = (d > 0.0f) ? rsqrtf(fmaxf(d, 1e-12f)) : 0.0f;
}

__global__ __launch_bounds__(256) void k_norm(const long long* __restrict__ ei,
                                              const float* __restrict__ w,
                                              const float* __restrict__ dinv,
                                              float* __restrict__ nrm) {
    int e = blockIdx.x * blockDim.x + threadIdx.x;
    if (e >= N_EDGES) return;
    int s = (int)ei[e];
    int d = (int)ei[N_EDGES + e];
    nrm[e] = dinv[s] * w[e] * dinv[d];
}

// ---------------------------------------------------------------------------
// WMMA GEMM: C[N_NODES x 64] = A[N_NODES x K] * B[K x 64], fp32 via
// v_wmma_f32_16x16x4_f32. One wave owns a 16-row block and all 4 column
// tiles (A fragment loaded once per k-step; B is 32KB, cache resident).
// A frag (16x4 f32): lanes 0-15 K={0,1}, lanes 16-31 K={2,3} (K = vgpr+2*half)
// B frag (4x16 f32): same K mapping, N = lane%16
// C/D (16x16 f32):   row = vgpr + 8*half, col = lane%16
// ---------------------------------------------------------------------------
template <int K>
__global__ __launch_bounds__(256) void k_gemm_wmma(const float* __restrict__ A,
                                                   const float* __restrict__ B,
                                                   float* __restrict__ C) {
    int wave = (int)((blockIdx.x * blockDim.x + threadIdx.x) >> 5);
    int lane = threadIdx.x & 31;
    if (wave >= N_TILES) return;           // wave-uniform: EXEC stays all-ones
    int half = lane >> 4;
    int lr   = lane & 15;
    int r0   = wave * 16;

    v8f acc0 = {}, acc1 = {}, acc2 = {}, acc3 = {};
    const float* arow = A + (size_t)(r0 + lr) * K;

    for (int k0 = 0; k0 < K; k0 += 4) {
        v2f a;
        a.x = arow[k0 + 2 * half];
        a.y = arow[k0 + 2 * half + 1];
        const float* brow0 = B + (size_t)(k0 + 2 * half) * D_HID;
        const float* brow1 = brow0 + D_HID;
        v2f b;
        b.x = brow0[lr];      b.y = brow1[lr];
        acc0 = __builtin_amdgcn_wmma_f32_16x16x4_f32(false, a, false, b, (short)0, acc0, false, false);
        b.x = brow0[16 + lr]; b.y = brow1[16 + lr];
        acc1 = __builtin_amdgcn_wmma_f32_16x16x4_f32(false, a, false, b, (short)0, acc1, false, false);
        b.x = brow0[32 + lr]; b.y = brow1[32 + lr];
        acc2 = __builtin_amdgcn_wmma_f32_16x16x4_f32(false, a, false, b, (short)0, acc2, false, false);
        b.x = brow0[48 + lr]; b.y = brow1[48 + lr];
        acc3 = __builtin_amdgcn_wmma_f32_16x16x4_f32(false, a, false, b, (short)0, acc3, false, false);
    }

#pragma unroll
    for (int v = 0; v < 8; ++v) {
        int row = r0 + v + 8 * half;
        float* crow = C + (size_t)row * D_HID + lr;
        crow[0]  = acc0[v];
        crow[16] = acc1[v];
        crow[32] = acc2[v];
        crow[48] = acc3[v];
    }
}

// ---------------------------------------------------------------------------
// out[i][c] = bias[c] + h[i][c] * dinv[i]^2   (bias + self-loop contribution)
// ---------------------------------------------------------------------------
__global__ __launch_bounds__(256) void k_seed(const float* __restrict__ h,
                                              const float* __restrict__ dinv,
                                              const float* __restrict__ bias,
                                              float* __restrict__ out) {
    int idx = blockIdx.x * blockDim.x + threadIdx.x;
    if (idx >= N_NODES * D_HID) return;
    int i = idx >> 6;
    int c = idx & 63;
    float di = dinv[i];
    out[idx] = bias[c] + h[idx] * di * di;
}

// ---------------------------------------------------------------------------
// Edge scatter: one wave per edge; lane covers 2 channels (float2 gather,
// two fp32 hardware atomics). Gathers/atomics land in the 192MB L2 since the
// per-layer feature matrix is only 12.8MB.
// ---------------------------------------------------------------------------
__global__ __launch_bounds__(256) void k_scatter(const long long* __restrict__ ei,
                                                 const float* __restrict__ nrm,
                                                 const float* __restrict__ h,
                                                 float* __restrict__ out) {
    int lane   = threadIdx.x & 31;
    int wave   = (int)((blockIdx.x * blockDim.x + threadIdx.x) >> 5);
    int nwaves = (int)((gridDim.x * blockDim.x) >> 5);
    for (int e = wave; e < N_EDGES; e += nwaves) {
        int   s = (int)ei[e];
        int   d = (int)ei[N_EDGES + e];
        float w = nrm[e];
        const float2* hp = (const float2*)(h + (size_t)s * D_HID) + lane;
        float2 v = *hp;
        float* op = out + (size_t)d * D_HID + 2 * lane;
        unsafeAtomicAdd(op,     v.x * w);
        unsafeAtomicAdd(op + 1, v.y * w);
    }
}

// ---------------------------------------------------------------------------
// BN stats: per-block LDS reduction of per-channel sum/sumsq, then atomics.
// ---------------------------------------------------------------------------
__global__ __launch_bounds__(256) void k_bn_stats(const float* __restrict__ h,
                                                  float* __restrict__ stats) {
    __shared__ float ssum[256];
    __shared__ float ssq[256];
    int t = threadIdx.x;
    int c = t & 63;
    int g = t >> 6;   // 4 row-groups per block
    float sum = 0.0f, sq = 0.0f;
    for (int i = blockIdx.x * 4 + g; i < N_NODES; i += gridDim.x * 4) {
        float v = h[(size_t)i * D_HID + c];
        sum += v;
        sq  += v * v;
    }
    ssum[t] = sum;
    ssq[t]  = sq;
    __syncthreads();
    if (g == 0) {
        sum = ssum[c] + ssum[64 + c] + ssum[128 + c] + ssum[192 + c];
        sq  = ssq[c]  + ssq[64 + c]  + ssq[128 + c]  + ssq[192 + c];
        unsafeAtomicAdd(&stats[c], sum);
        unsafeAtomicAdd(&stats[64 + c], sq);
    }
}

// ---------------------------------------------------------------------------
// BN (biased var, batch stats) + ReLU, elementwise (safe in place).
// ---------------------------------------------------------------------------
__global__ __launch_bounds__(256) void k_bn_apply(const float* __restrict__ stats,
                                                  const float* __restrict__ gamma,
                                                  const float* __restrict__ beta,
                                                  const float* __restrict__ in,
                                                  float* __restrict__ out) {
    int idx = blockIdx.x * blockDim.x + threadIdx.x;
    if (idx >= N_NODES * D_HID) return;
    int c = idx & 63;
    const float invn = 1.0f / (float)N_NODES;
    float mean  = stats[c] * invn;
    float var   = stats[64 + c] * invn - mean * mean;
    float scale = gamma[c] * rsqrtf(var + BN_EPS);
    float v = (in[idx] - mean) * scale + beta[c];
    out[idx] = fmaxf(v, 0.0f);
}

// ---------------------------------------------------------------------------
extern "C" void kernel_launch(void* const* d_in, const int* in_sizes, int n_in,
                              void* d_out, int out_size, void* d_ws, size_t ws_size,
                              hipStream_t stream) {
    const float*     x      = (const float*)d_in[0];
    const long long* ei_sc  = (const long long*)d_in[1];   // [2,E] int64
    const float*     wsc    = (const float*)d_in[2];
    // d_in[3] edge_index_fc: unused by the reference computation
    const float*     wfc    = (const float*)d_in[4];
    const float*     alpha  = (const float*)d_in[5];
    const float*     W1     = (const float*)d_in[6];
    const float*     b1     = (const float*)d_in[7];
    const float*     W2     = (const float*)d_in[8];
    const float*     b2     = (const float*)d_in[9];
    const float*     gamma1 = (const float*)d_in[10];
    const float*     beta1  = (const float*)d_in[11];
    const float*     gamma2 = (const float*)d_in[12];
    const float*     beta2  = (const float*)d_in[13];
    float* out = (float*)d_out;

    // workspace layout (floats)
    float* ws    = (float*)d_ws;
    float* w     = ws;                       // E
    float* nrm   = w    + N_EDGES;           // E
    float* deg   = nrm  + N_EDGES;           // N
    float* dinv  = deg  + N_NODES;           // N
    float* h     = dinv + N_NODES;           // N*64  (GEMM out)
    float* buf   = h    + N_NODES * D_HID;   // N*64  (aggregate / BN)
    float* stats = buf  + N_NODES * D_HID;   // 256

    const int T = 256;
    const int gN   = (N_NODES + T - 1) / T;           // 196
    const int gE   = (N_EDGES + T - 1) / T;           // 6250
    const int gNC  = (N_NODES * D_HID + T - 1) / T;   // 12500
    const int gMM  = (N_TILES + 7) / 8;               // 8 waves/block
    const int gSC  = 1024;                            // grid-stride scatter
    const int gST  = 512;

    k_init<<<gN, T, 0, stream>>>(deg, stats);
    k_mix_deg<<<gE, T, 0, stream>>>(wsc, wfc, alpha, ei_sc + N_EDGES, w, deg);
    k_dinv<<<gN, T, 0, stream>>>(deg, dinv);
    k_norm<<<gE, T, 0, stream>>>(ei_sc, w, dinv, nrm);

    // ---- layer 1 ----
    k_gemm_wmma<D_INF><<<gMM, T, 0, stream>>>(x, W1, h);
    k_seed<<<gNC, T, 0, stream>>>(h, dinv, b1, buf);
    k_scatter<<<gSC, T, 0, stream>>>(ei_sc, nrm, h, buf);
    k_bn_stats<<<gST, T, 0, stream>>>(buf, stats);
    k_bn_apply<<<gNC, T, 0, stream>>>(stats, gamma1, beta1, buf, buf);

    // ---- layer 2 ----
    k_gemm_wmma<D_HID><<<gMM, T, 0, stream>>>(buf, W2, h);
    k_seed<<<gNC, T, 0, stream>>>(h, dinv, b2, out);
    k_scatter<<<gSC, T, 0, stream>>>(ei_sc, nrm, h, out);
    k_bn_stats<<<gST, T, 0, stream>>>(out, stats + 128);
    k_bn_apply<<<gNC, T, 0, stream>>>(stats + 128, gamma2, beta2, out, out);
}